// Choquet_Integral_18236431138960
// MI455X (gfx1250) — compile-verified
//
#include <hip/hip_runtime.h>
#include <hip/hip_bf16.h>
#include <stdint.h>

typedef float v2f __attribute__((ext_vector_type(2)));
typedef float v8f __attribute__((ext_vector_type(8)));

#define N_    2048
#define S_    8
#define D_    512
#define H_    8
#define VAR_  255        // 2^S - 1 lattice rows
#define LN_EPS 1e-5f

// descending compare-exchange with index tracking (compiler -> v_cndmask, branchless)
#define CSWAP(i, j)                                            \
  do {                                                         \
    if (v[i] < v[j]) {                                         \
      float _t = v[i]; v[i] = v[j]; v[j] = _t;                 \
      int _ti = ix[i]; ix[i] = ix[j]; ix[j] = _ti;             \
    }                                                          \
  } while (0)

__global__ __launch_bounds__(512) void choquet_ln_prelu_kernel(
    const float* __restrict__ x,       // (N, S, D)
    const float* __restrict__ FMg,     // (255, H)
    const float* __restrict__ lnw,     // (H, D)
    const float* __restrict__ lnb,     // (H, D)
    const float* __restrict__ prelu,   // (1,)
    float* __restrict__ out)           // (N, H, D)
{
  __shared__ __align__(16) float fm[VAR_ * H_];   // 8160 B, gathered by all threads
  __shared__ float rs[512];
  __shared__ float rq[512];
  __shared__ float stats[2];                      // mean, rstd

  const int tid = threadIdx.x;       // = d column
  const int n   = blockIdx.x;
  const int d   = tid;

  // ---- Stage FM into LDS with CDNA5 async global->LDS DMA (510 lanes x 16B) ----
  if (tid < (VAR_ * H_) / 4) {
    uint32_t ldsoff = (uint32_t)(uintptr_t)(&fm[0]) + (uint32_t)(tid * 16);
    uint32_t goff   = (uint32_t)(tid * 16);
    asm volatile("global_load_async_to_lds_b128 %0, %1, %2"
                 :: "v"(ldsoff), "v"(goff), "s"(FMg) : "memory");
  }

  // ---- Load this thread's 8-element source column (coalesced per s), overlapping the DMA ----
  float v[8]; int ix[8];
  const float* xp = x + (size_t)n * (S_ * D_) + d;
#pragma unroll
  for (int s = 0; s < 8; ++s) { v[s] = xp[s * D_]; ix[s] = s; }

  // ---- Descending sort: Batcher odd-even mergesort, 19 comparators ----
  CSWAP(0, 1); CSWAP(2, 3); CSWAP(4, 5); CSWAP(6, 7);
  CSWAP(0, 2); CSWAP(1, 3); CSWAP(4, 6); CSWAP(5, 7);
  CSWAP(1, 2); CSWAP(5, 6);
  CSWAP(0, 4); CSWAP(1, 5); CSWAP(2, 6); CSWAP(3, 7);
  CSWAP(2, 4); CSWAP(3, 5);
  CSWAP(1, 2); CSWAP(3, 4); CSWAP(5, 6);

  // Wait for this wave's async DMA, then make FM visible block-wide
  asm volatile("s_wait_asynccnt 0" ::: "memory");
  __syncthreads();

  // ---- Choquet accumulation: y[h] = sum_s diff_s * FM[latt_s, h] ----
  float y0 = 0.f, y1 = 0.f, y2 = 0.f, y3 = 0.f, y4 = 0.f, y5 = 0.f, y6 = 0.f, y7 = 0.f;
  int am = 0;
#pragma unroll
  for (int s = 0; s < 8; ++s) {
    am += (1 << ix[s]);
    float dd = (s < 7) ? (v[s] - v[s + 1]) : v[7];
    const float4* rp = (const float4*)&fm[(am - 1) * H_];  // 32B-aligned row
    float4 r0 = rp[0], r1 = rp[1];
    y0 = fmaf(dd, r0.x, y0); y1 = fmaf(dd, r0.y, y1);
    y2 = fmaf(dd, r0.z, y2); y3 = fmaf(dd, r0.w, y3);
    y4 = fmaf(dd, r1.x, y4); y5 = fmaf(dd, r1.y, y5);
    y6 = fmaf(dd, r1.z, y6); y7 = fmaf(dd, r1.w, y7);
  }

  // ---- Block-wide mean / E[y^2] over H*D = 4096 values ----
  float ls = y0 + y1 + y2 + y3 + y4 + y5 + y6 + y7;
  float lq = y0*y0 + y1*y1 + y2*y2 + y3*y3 + y4*y4 + y5*y5 + y6*y6 + y7*y7;
  rs[tid] = ls; rq[tid] = lq;
  __syncthreads();
#pragma unroll
  for (int off = 256; off >= 64; off >>= 1) {
    if (tid < off) { rs[tid] += rs[tid + off]; rq[tid] += rq[tid + off]; }
    __syncthreads();
  }

  // ---- Tail reduction of 64 partials via v_wmma_f32_16x16x4_f32 (B = ones), wave 0 only ----
  if (tid < 32) {
    v2f as, aq, bo;
    as.x = rs[tid * 2]; as.y = rs[tid * 2 + 1];
    aq.x = rq[tid * 2]; aq.y = rq[tid * 2 + 1];
    bo.x = 1.0f;        bo.y = 1.0f;
    v8f c0 = {}; v8f c1 = {};
    c0 = __builtin_amdgcn_wmma_f32_16x16x4_f32(false, as, false, bo, (short)0, c0, false, false);
    c1 = __builtin_amdgcn_wmma_f32_16x16x4_f32(false, aq, false, bo, (short)0, c1, false, false);
    // per lane: sum 8 C rows -> half-total; lanes 0-15 hold rows 0-7, lanes 16-31 rows 8-15
    float ts = c0[0] + c0[1] + c0[2] + c0[3] + c0[4] + c0[5] + c0[6] + c0[7];
    float tq = c1[0] + c1[1] + c1[2] + c1[3] + c1[4] + c1[5] + c1[6] + c1[7];
    ts += __shfl_xor(ts, 16);
    tq += __shfl_xor(tq, 16);
    if (tid == 0) {
      float mean = ts * (1.0f / (H_ * D_));
      float var  = tq * (1.0f / (H_ * D_)) - mean * mean;
      stats[0] = mean;
      stats[1] = rsqrtf(var + LN_EPS);
    }
  }
  __syncthreads();

  // ---- Normalize + affine + PReLU, coalesced stores ----
  const float mean = stats[0];
  const float rstd = stats[1];
  const float p    = prelu[0];
  float yy[8] = { y0, y1, y2, y3, y4, y5, y6, y7 };
  float* op = out + (size_t)n * (H_ * D_) + d;
#pragma unroll
  for (int h = 0; h < H_; ++h) {
    float val = (yy[h] - mean) * rstd * lnw[h * D_ + d] + lnb[h * D_ + d];
    val = (val > 0.0f) ? val : p * val;
    op[h * D_] = val;
  }
}

extern "C" void kernel_launch(void* const* d_in, const int* in_sizes, int n_in,
                              void* d_out, int out_size, void* d_ws, size_t ws_size,
                              hipStream_t stream) {
  const float* x   = (const float*)d_in[0];
  const float* FMg = (const float*)d_in[1];
  const float* lnw = (const float*)d_in[2];
  const float* lnb = (const float*)d_in[3];
  const float* pw  = (const float*)d_in[4];
  float* out = (float*)d_out;
  (void)in_sizes; (void)n_in; (void)out_size; (void)d_ws; (void)ws_size;
  choquet_ln_prelu_kernel<<<dim3(N_), dim3(512), 0, stream>>>(x, FMg, lnw, lnb, pw, out);
}